// AttnDecoderRNN_16544214024590
// MI455X (gfx1250) — compile-verified
//
#include <hip/hip_runtime.h>
#include <math.h>

// Problem constants (from reference)
#define HID   1024
#define VOC   50257
#define LATT  40
#define TWOH  2048
#define MAXK  3072   // largest GEMV depth (attn/comb use 3H)

typedef __attribute__((ext_vector_type(2))) float v2f;
typedef __attribute__((ext_vector_type(8))) float v8f;

// ---------------------------------------------------------------------------
// Generic f32 GEMV via V_WMMA_F32_16X16X4_F32.
// y[r] = relu?( dot(W[r,:], x) + bias[r] + acc[r] )
// One wave computes 16 output rows. A = 16x4 tile of W (streamed from HBM,
// optionally non-temporal), B = x broadcast across the 16 N-columns, fed
// from LDS (conflict-free broadcast ds_load_b64). EXEC stays all-1s: row
// index clamped for loads, stores predicated on the real row range.
//
// A layout (ISA 7.12.2, 32-bit A 16x4): lane m in 0..15 -> (K=0,K=1),
// lane m+16 -> (K=2,K=3). B 4x16: lanes 0..15 carry K=0/1 in VGPR0/1,
// lanes 16..31 carry K=2/3. D: VGPR r -> M=r (lanes 0..15), M=8+r (16..31).
//
// NT=1 marks the weight stream non-temporal (TH_NT): used for the 206 MB
// out_w read so the ~81 MB of recurrent weights stay resident in the 192 MB
// global L2 across graph replays.
// ---------------------------------------------------------------------------
template <int NT>
__global__ void wmma_gemv_f32(const float* __restrict__ W,
                              const float* __restrict__ x,
                              const float* __restrict__ bias,  // may be null
                              const float* acc,                // may be null / alias y
                              float* y,
                              int rows, int cols, int relu)
{
  __shared__ float xs[MAXK];

  // Cooperative stage of x into LDS (before any wave-uniform exit so the
  // barrier is well-formed for the whole workgroup).
  for (int j = threadIdx.x; j < cols; j += blockDim.x) xs[j] = x[j];
  __syncthreads();

  const int lane  = threadIdx.x & 31;
  const int wave  = (blockIdx.x * blockDim.x + threadIdx.x) >> 5;
  const int r0    = wave * 16;
  if (r0 >= rows) return;                       // wave-uniform exit

  const int m     = lane & 15;
  const int khalf = (lane >> 4) << 1;           // 0 for lanes 0-15, 2 for 16-31
  int row = r0 + m;
  if (row >= rows) row = rows - 1;              // clamp: keep EXEC all-1s

  const float* wp = W + (size_t)row * (size_t)cols + khalf;
  const float* xp = xs + khalf;

  v8f c = {0.f, 0.f, 0.f, 0.f, 0.f, 0.f, 0.f, 0.f};
  for (int k = 0; k < cols; k += 4) {
    v2f a;
    if (NT) a = __builtin_nontemporal_load((const v2f*)(wp + k));
    else    a = *(const v2f*)(wp + k);          // W[row][k+khalf .. +1]
    v2f b = *(const v2f*)(xp + k);              // LDS broadcast
    c = __builtin_amdgcn_wmma_f32_16x16x4_f32(
            /*neg_a=*/false, a, /*neg_b=*/false, b,
            /*c_mod=*/(short)0, c, /*reuse_a=*/false, /*reuse_b=*/false);
  }

  // All 16 N-columns of D are identical; lanes with N==0 (lane 0 and 16)
  // own M = r and M = 8+r respectively across the 8 accumulator VGPRs.
  if ((lane & 15) == 0) {
    const int mbase = r0 + ((lane >> 4) << 3);
#pragma unroll
    for (int r = 0; r < 8; ++r) {
      int rr = mbase + r;
      if (rr < rows) {
        float v = c[r];
        if (bias) v += bias[rr];
        if (acc)  v += acc[rr];
        if (relu) v = v > 0.f ? v : 0.f;
        y[rr] = v;
      }
    }
  }
}

// ---------------------------------------------------------------------------
// attn_in = [ emb[token] ; h0.flatten() ]  (3H) ; also xcat[0:H] = embedded
// ---------------------------------------------------------------------------
__global__ void prep_kernel(const int* __restrict__ tok,
                            const float* __restrict__ emb,
                            const float* __restrict__ h0,
                            float* __restrict__ attn_in,
                            float* __restrict__ xcat)
{
  int j = blockIdx.x * blockDim.x + threadIdx.x;
  if (j < HID) {
    float e = emb[(size_t)tok[0] * HID + j];
    attn_in[j] = e;
    xcat[j]    = e;
  } else if (j < 3 * HID) {
    attn_in[j] = h0[j - HID];                   // h0 is [2,1,H] contiguous
  }
}

// ---------------------------------------------------------------------------
// softmax over 40 logits, single wave32 (each lane covers l and l+32)
// ---------------------------------------------------------------------------
__global__ void softmax40_kernel(const float* __restrict__ z,
                                 float* __restrict__ w40,
                                 float* __restrict__ aw_out)
{
  int t = threadIdx.x;                          // 32 threads
  float z0 = (t      < LATT) ? z[t]      : -INFINITY;
  float z1 = (t + 32 < LATT) ? z[t + 32] : -INFINITY;
  float mx = fmaxf(z0, z1);
  for (int o = 16; o > 0; o >>= 1) mx = fmaxf(mx, __shfl_xor(mx, o));
  float e0 = (t      < LATT) ? expf(z0 - mx) : 0.f;
  float e1 = (t + 32 < LATT) ? expf(z1 - mx) : 0.f;
  float s = e0 + e1;
  for (int o = 16; o > 0; o >>= 1) s += __shfl_xor(s, o);
  float inv = 1.f / s;
  if (t < LATT)      { w40[t]      = e0 * inv; aw_out[t]      = e0 * inv; }
  if (t + 32 < LATT) { w40[t + 32] = e1 * inv; aw_out[t + 32] = e1 * inv; }
}

// ---------------------------------------------------------------------------
// attn_applied[j] = sum_l w40[l] * enc[l][j]   (j < 2H), written into xcat+H
// ---------------------------------------------------------------------------
__global__ void attn_apply_kernel(const float* __restrict__ w40,
                                  const float* __restrict__ enc,
                                  float* __restrict__ dst)
{
  int j = blockIdx.x * blockDim.x + threadIdx.x;
  if (j < TWOH) {
    float s = 0.f;
#pragma unroll 8
    for (int l = 0; l < LATT; ++l) s = fmaf(w40[l], enc[l * TWOH + j], s);
    dst[j] = s;
  }
}

// ---------------------------------------------------------------------------
// LSTM elementwise: gates g[4H] (i,f,g,o) -> h2, c2
// ---------------------------------------------------------------------------
__device__ __forceinline__ float sigmoidf(float x) { return 1.f / (1.f + expf(-x)); }

__global__ void lstm_elem_kernel(const float* __restrict__ g,
                                 const float* __restrict__ cprev,
                                 float* __restrict__ h_ws,
                                 float* __restrict__ h_out,
                                 float* __restrict__ c_out)
{
  int j = blockIdx.x * blockDim.x + threadIdx.x;
  if (j < HID) {
    float ig = sigmoidf(g[j]);
    float fg = sigmoidf(g[HID + j]);
    float gg = tanhf(g[2 * HID + j]);
    float og = sigmoidf(g[3 * HID + j]);
    float c2 = fg * cprev[j] + ig * gg;
    float h2 = og * tanhf(c2);
    h_ws[j] = h2; h_out[j] = h2; c_out[j] = c2;
  }
}

// ---------------------------------------------------------------------------
// log-sum-exp stats over V logits (single 1024-thread block; z is L2-resident)
// ---------------------------------------------------------------------------
__global__ void lse_kernel(const float* __restrict__ z, float* __restrict__ lse, int n)
{
  __shared__ float sm[32];
  int t = threadIdx.x;
  float m = -INFINITY;
  for (int v = t; v < n; v += blockDim.x) m = fmaxf(m, z[v]);
  for (int o = 16; o > 0; o >>= 1) m = fmaxf(m, __shfl_xor(m, o));
  if ((t & 31) == 0) sm[t >> 5] = m;
  __syncthreads();
  float M = sm[t & 31];                          // each wave re-reduces full sm
  for (int o = 16; o > 0; o >>= 1) M = fmaxf(M, __shfl_xor(M, o));
  __syncthreads();
  float s = 0.f;
  for (int v = t; v < n; v += blockDim.x) s += expf(z[v] - M);
  for (int o = 16; o > 0; o >>= 1) s += __shfl_xor(s, o);
  if ((t & 31) == 0) sm[t >> 5] = s;
  __syncthreads();
  if (t < 32) {
    float ss = sm[t];
    for (int o = 16; o > 0; o >>= 1) ss += __shfl_xor(ss, o);
    if (t == 0) lse[0] = M + logf(ss);
  }
}

__global__ void finalize_kernel(const float* __restrict__ z,
                                const float* __restrict__ lse,
                                float* __restrict__ lp, int n)
{
  int v = blockIdx.x * blockDim.x + threadIdx.x;
  if (v < n) lp[v] = z[v] - lse[0];
}

// ---------------------------------------------------------------------------
extern "C" void kernel_launch(void* const* d_in, const int* in_sizes, int n_in,
                              void* d_out, int out_size, void* d_ws, size_t ws_size,
                              hipStream_t stream)
{
  (void)in_sizes; (void)n_in; (void)out_size; (void)ws_size;

  const int*   tok    = (const int*)  d_in[0];
  const float* h0     = (const float*)d_in[1];
  const float* c0     = (const float*)d_in[2];
  const float* enc    = (const float*)d_in[3];
  const float* emb    = (const float*)d_in[4];
  const float* attn_w = (const float*)d_in[5];
  const float* attn_b = (const float*)d_in[6];
  const float* comb_w = (const float*)d_in[7];
  const float* comb_b = (const float*)d_in[8];
  const float* w_ih0  = (const float*)d_in[9];
  const float* w_hh0  = (const float*)d_in[10];
  const float* b_ih0  = (const float*)d_in[11];
  const float* b_hh0  = (const float*)d_in[12];
  const float* w_ih1  = (const float*)d_in[13];
  const float* w_hh1  = (const float*)d_in[14];
  const float* b_ih1  = (const float*)d_in[15];
  const float* b_hh1  = (const float*)d_in[16];
  const float* out_w  = (const float*)d_in[17];
  const float* out_b  = (const float*)d_in[18];

  float* ws      = (float*)d_ws;
  float* attn_in = ws;                 // 3072
  float* xcat    = ws + 3072;          // 3072  [embedded ; attn_applied]
  float* l40     = ws + 6144;          // 64
  float* w40     = ws + 6208;          // 64
  float* xvec    = ws + 6272;          // 1024
  float* gbuf    = ws + 7296;          // 4096  (reused by both LSTM layers)
  float* hl0     = ws + 11392;         // 1024
  float* hl1     = ws + 12416;         // 1024
  float* zv      = ws + 13440;         // 50304
  float* lse     = ws + 63744;         // 1

  float* out   = (float*)d_out;
  float* lp    = out;                  // [V]     log_probs
  float* hout  = out + VOC;            // [2,H]   h_new
  float* cout  = out + VOC + TWOH;     // [2,H]   c_new
  float* awout = out + VOC + 2*TWOH;   // [40]    attn_weights

  auto gemv_blocks = [](int rows) {
    int waves = (rows + 15) / 16;
    return (waves * 32 + 255) / 256;
  };

  // 1. embedded + attn_in
  prep_kernel<<<(3 * HID + 255) / 256, 256, 0, stream>>>(tok, emb, h0, attn_in, xcat);
  // 2. attention logits (40 x 3072)
  wmma_gemv_f32<0><<<gemv_blocks(LATT), 256, 0, stream>>>(attn_w, attn_in, attn_b, nullptr,
                                                          l40, LATT, 3 * HID, 0);
  // 3. softmax(40) -> attn weights (also to d_out)
  softmax40_kernel<<<1, 32, 0, stream>>>(l40, w40, awout);
  // 4. attn_applied -> xcat[H : 3H]
  attn_apply_kernel<<<(TWOH + 255) / 256, 256, 0, stream>>>(w40, enc, xcat + HID);
  // 5. x = relu(comb_w @ xcat + comb_b)   (1024 x 3072)
  wmma_gemv_f32<0><<<gemv_blocks(HID), 256, 0, stream>>>(comb_w, xcat, comb_b, nullptr,
                                                         xvec, HID, 3 * HID, 1);
  // 6. LSTM layer 0 gates: g = W_ih0@x + b_ih0 + b_hh0 ; g += W_hh0@h0[0]
  wmma_gemv_f32<0><<<gemv_blocks(4 * HID), 256, 0, stream>>>(w_ih0, xvec, b_ih0, b_hh0,
                                                             gbuf, 4 * HID, HID, 0);
  wmma_gemv_f32<0><<<gemv_blocks(4 * HID), 256, 0, stream>>>(w_hh0, h0, nullptr, gbuf,
                                                             gbuf, 4 * HID, HID, 0);
  lstm_elem_kernel<<<(HID + 255) / 256, 256, 0, stream>>>(gbuf, c0, hl0, hout, cout);
  // 7. LSTM layer 1
  wmma_gemv_f32<0><<<gemv_blocks(4 * HID), 256, 0, stream>>>(w_ih1, hl0, b_ih1, b_hh1,
                                                             gbuf, 4 * HID, HID, 0);
  wmma_gemv_f32<0><<<gemv_blocks(4 * HID), 256, 0, stream>>>(w_hh1, h0 + HID, nullptr, gbuf,
                                                             gbuf, 4 * HID, HID, 0);
  lstm_elem_kernel<<<(HID + 255) / 256, 256, 0, stream>>>(gbuf, c0 + HID, hl1,
                                                          hout + HID, cout + HID);
  // 8. vocab projection (50257 x 1024): 206 MB streamed NON-TEMPORAL so the
  //    recurrent weights (~81 MB) stay resident in the 192 MB global L2.
  wmma_gemv_f32<1><<<gemv_blocks(VOC), 256, 0, stream>>>(out_w, hl1, out_b, nullptr,
                                                         zv, VOC, HID, 0);
  // 9. log-softmax
  lse_kernel<<<1, 1024, 0, stream>>>(zv, lse, VOC);
  finalize_kernel<<<(VOC + 255) / 256, 256, 0, stream>>>(zv, lse, lp, VOC);
}